// SGFormer_60610578481398
// MI455X (gfx1250) — compile-verified
//
#include <hip/hip_runtime.h>

// SGFormer forward on MI455X (gfx1250), fp32 end-to-end.
// Matrix work runs on V_WMMA_F32_16X16X4_F32 (wave32 matrix pipe).
// N*D f32 tables (100 MB) are L2-resident (192 MB), so edge gather/scatter
// and GEMM A-streams are L2-bound; total HBM traffic ~3 GB -> ~0.13 ms floor.
// GEMM inner loop: pointer-increment addressing (no per-iter u64 muls),
// 2 M-tiles x 4 N-tiles per wave (8 WMMAs per 12 loads).

#define N_NODES 100000
#define N_EDGES 1600000
#define DIM     256
#define OUTD    128
#define EPSV    1e-5f

typedef float v2f __attribute__((ext_vector_type(2)));
typedef float v8f __attribute__((ext_vector_type(8)));

// ---------------------------------------------------------------- utilities
__device__ __forceinline__ float block_reduce_sum(float val, float* red) {
  const int t = threadIdx.x;
  red[t] = val;
  __syncthreads();
  for (int s = 128; s > 0; s >>= 1) {
    if (t < s) red[t] += red[t + s];
    __syncthreads();
  }
  float r = red[0];
  __syncthreads();
  return r;
}

__global__ __launch_bounds__(256) void zero_kernel(float* __restrict__ p, size_t n) {
  size_t i = (size_t)blockIdx.x * blockDim.x + threadIdx.x;
  const size_t stride = (size_t)gridDim.x * blockDim.x;
  for (; i < n; i += stride) p[i] = 0.0f;
}

// ------------------------------------------------------------- WMMA GEMM f32
// C[M x Nn] = A (strided: rsA/csA) @ B[K x Nn, row-major ldB] (+ bias).
// Block = 8 waves; each wave owns a 32(M) x 64(N) tile: 2 M-tiles x 4 N-tiles
// = 8 f32 accumulators (64 VGPRs). Per k-step (K+=4): 4 A loads + 8 B loads
// feed 8 WMMAs; all addresses advance by pointer increment (no u64 muls).
// K must be a multiple of 4 (256 and 100000 both are). Nn = 64 * gridDim.x.
// Wave-uniform early-exit keeps EXEC all-ones across every WMMA (ISA §7.12).
__global__ __launch_bounds__(256)
void wmma_gemm_f32(const float* __restrict__ A, int rsA, int csA,
                   const float* __restrict__ B, int ldB,
                   float* __restrict__ C, int ldC,
                   const float* __restrict__ bias,
                   int M, int K)
{
  const int lane = threadIdx.x & 31;
  const int wave = threadIdx.x >> 5;
  const int m0 = (blockIdx.y * 8 + wave) * 32;   // two 16-row tiles: m0, m0+16
  if (m0 >= M) return;                           // wave-uniform
  const int n0 = blockIdx.x * 64;
  const int half = lane >> 4;                    // lanes 16-31: K+2/K+3 (A), rows M+8 (C)
  const int ml = lane & 15;

  v8f acc00 = {}; v8f acc01 = {}; v8f acc02 = {}; v8f acc03 = {};
  v8f acc10 = {}; v8f acc11 = {}; v8f acc12 = {}; v8f acc13 = {};

  int arow0 = m0 + ml;      if (arow0 >= M) arow0 = M - 1;  // clamped rows never stored
  int arow1 = m0 + 16 + ml; if (arow1 >= M) arow1 = M - 1;

  const size_t csAs = (size_t)csA;
  const size_t ldBs = (size_t)ldB;
  const float* __restrict__ Ap = A + (size_t)arow0 * rsA + (size_t)(2 * half) * csAs;
  const float* __restrict__ Aq = A + (size_t)arow1 * rsA + (size_t)(2 * half) * csAs;
  const float* __restrict__ Bp = B + (size_t)(2 * half) * ldBs + n0 + ml;
  const size_t aStep = 4 * csAs;
  const size_t bStep = 4 * ldBs;

  for (int k0 = 0; k0 < K; k0 += 4) {
    v2f a0; a0.x = Ap[0];  a0.y = Ap[csAs];
    v2f a1; a1.x = Aq[0];  a1.y = Aq[csAs];
    v2f b0; b0.x = Bp[0];  b0.y = Bp[ldBs];
    v2f b1; b1.x = Bp[16]; b1.y = Bp[16 + ldBs];
    v2f b2; b2.x = Bp[32]; b2.y = Bp[32 + ldBs];
    v2f b3; b3.x = Bp[48]; b3.y = Bp[48 + ldBs];
    Ap += aStep; Aq += aStep; Bp += bStep;
    acc00 = __builtin_amdgcn_wmma_f32_16x16x4_f32(false, a0, false, b0, (short)0, acc00, false, false);
    acc01 = __builtin_amdgcn_wmma_f32_16x16x4_f32(false, a0, false, b1, (short)0, acc01, false, false);
    acc02 = __builtin_amdgcn_wmma_f32_16x16x4_f32(false, a0, false, b2, (short)0, acc02, false, false);
    acc03 = __builtin_amdgcn_wmma_f32_16x16x4_f32(false, a0, false, b3, (short)0, acc03, false, false);
    acc10 = __builtin_amdgcn_wmma_f32_16x16x4_f32(false, a1, false, b0, (short)0, acc10, false, false);
    acc11 = __builtin_amdgcn_wmma_f32_16x16x4_f32(false, a1, false, b1, (short)0, acc11, false, false);
    acc12 = __builtin_amdgcn_wmma_f32_16x16x4_f32(false, a1, false, b2, (short)0, acc12, false, false);
    acc13 = __builtin_amdgcn_wmma_f32_16x16x4_f32(false, a1, false, b3, (short)0, acc13, false, false);
  }

  const float bv0 = bias ? bias[n0 + ml]      : 0.0f;
  const float bv1 = bias ? bias[n0 + 16 + ml] : 0.0f;
  const float bv2 = bias ? bias[n0 + 32 + ml] : 0.0f;
  const float bv3 = bias ? bias[n0 + 48 + ml] : 0.0f;
  for (int r = 0; r < 8; ++r) {
    const int row0 = m0 + half * 8 + r;        // C VGPR r -> rows r / r+8
    if (row0 < M) {
      const size_t base = (size_t)row0 * ldC + n0 + ml;
      C[base]      = acc00[r] + bv0;
      C[base + 16] = acc01[r] + bv1;
      C[base + 32] = acc02[r] + bv2;
      C[base + 48] = acc03[r] + bv3;
    }
    const int row1 = row0 + 16;
    if (row1 < M) {
      const size_t base = (size_t)row1 * ldC + n0 + ml;
      C[base]      = acc10[r] + bv0;
      C[base + 16] = acc11[r] + bv1;
      C[base + 32] = acc12[r] + bv2;
      C[base + 48] = acc13[r] + bv3;
    }
  }
}

// --------------------------------------------------------------- row kernels
__global__ __launch_bounds__(256)
void ln_relu_kernel(const float* __restrict__ in, const float* __restrict__ g,
                    const float* __restrict__ b, float* __restrict__ out) {
  __shared__ float red[256];
  const int t = threadIdx.x;
  const size_t idx = (size_t)blockIdx.x * DIM + t;
  const float v = in[idx];
  const float mean = block_reduce_sum(v, red) * (1.0f / DIM);
  const float d = v - mean;
  const float var = block_reduce_sum(d * d, red) * (1.0f / DIM);
  const float y = d * rsqrtf(var + EPSV) * g[t] + b[t];
  out[idx] = fmaxf(y, 0.0f);
}

__global__ __launch_bounds__(256)
void rownorm_kernel(float* __restrict__ p) {
  __shared__ float red[256];
  const int t = threadIdx.x;
  const size_t idx = (size_t)blockIdx.x * DIM + t;
  const float v = p[idx];
  const float ss = block_reduce_sum(v * v, red);
  p[idx] = v * rsqrtf(ss);
}

// attn = (numq + nf*vsum) / (q.ksum + 2nf); x1 = relu(LN((attn + h)/2))
__global__ __launch_bounds__(256)
void attn_ln_kernel(const float* __restrict__ numq, const float* __restrict__ q,
                    const float* __restrict__ ksum, const float* __restrict__ vsum,
                    const float* __restrict__ h, const float* __restrict__ g,
                    const float* __restrict__ b, float* __restrict__ x1, float nf) {
  __shared__ float red[256];
  const int t = threadIdx.x;
  const size_t idx = (size_t)blockIdx.x * DIM + t;
  const float qv = q[idx];
  const float qdot = block_reduce_sum(qv * ksum[t], red);
  const float a = (numq[idx] + nf * vsum[t]) / (qdot + 2.0f * nf);
  const float tt = (a + h[idx]) * 0.5f;
  const float mean = block_reduce_sum(tt, red) * (1.0f / DIM);
  const float d = tt - mean;
  const float var = block_reduce_sum(d * d, red) * (1.0f / DIM);
  const float y = d * rsqrtf(var + EPSV) * g[t] + b[t];
  x1[idx] = fmaxf(y, 0.0f);
}

// ------------------------------------------------------ column stats (BN etc)
__global__ __launch_bounds__(256)
void col_stats_kernel(const float* __restrict__ in, float* __restrict__ sum,
                      float* __restrict__ sumsq, int M, int chunk) {
  const int t = threadIdx.x;
  const int r0 = blockIdx.x * chunk;
  const int r1 = min(M, r0 + chunk);
  float s = 0.0f, s2 = 0.0f;
  for (int r = r0; r < r1; ++r) {
    const float v = in[(size_t)r * DIM + t];
    s += v; s2 += v * v;
  }
  atomicAdd(&sum[t], s);
  if (sumsq) atomicAdd(&sumsq[t], s2);
}

__global__ __launch_bounds__(256)
void bn_relu_kernel(const float* __restrict__ in, const float* __restrict__ sum,
                    const float* __restrict__ sq, const float* __restrict__ g,
                    const float* __restrict__ b, const float* __restrict__ res,
                    float* __restrict__ out, float invM, size_t total) {
  const size_t idx = (size_t)blockIdx.x * 256 + threadIdx.x;
  if (idx >= total) return;
  const int c = (int)(idx & (DIM - 1));
  const float m = sum[c] * invM;
  const float var = sq[c] * invM - m * m;
  float y = (in[idx] - m) * rsqrtf(var + EPSV) * g[c] + b[c];
  y = fmaxf(y, 0.0f);
  if (res) y += res[idx];
  out[idx] = y;
}

// ---------------------------------------------------------------- edge phase
__global__ __launch_bounds__(256)
void deg_kernel(const int* __restrict__ ei, float* __restrict__ deg, int E) {
  const int e = blockIdx.x * 256 + threadIdx.x;
  if (e < E) atomicAdd(&deg[ei[E + e]], 1.0f);   // col = edge_index[1]
}

__global__ __launch_bounds__(256)
void edgew_kernel(const int* __restrict__ ei, const float* __restrict__ deg,
                  float* __restrict__ w, int E) {
  const int e = blockIdx.x * 256 + threadIdx.x;
  if (e < E) {
    const int r = ei[e], c = ei[E + e];
    const float dc = deg[c], dr = deg[r];
    const float ic = dc > 0.0f ? 1.0f / dc : 0.0f;
    const float ir = dr > 0.0f ? 1.0f / dr : 0.0f;
    w[e] = sqrtf(ic) * sqrtf(ir);
  }
}

// agg[col,:] += w[e] * g0[row,:]  (L2-resident tables, f32 atomics)
__global__ __launch_bounds__(256)
void scatter_kernel(const int* __restrict__ ei, const float* __restrict__ w,
                    const float* __restrict__ g0, float* __restrict__ agg, int E) {
  const size_t idx = (size_t)blockIdx.x * 256 + threadIdx.x;
  if (idx >= (size_t)E * DIM) return;
  const int e = (int)(idx >> 8);
  const int d = (int)(idx & (DIM - 1));
  const float we = w[e];
  if (we != 0.0f) {
    const int r = ei[e], c = ei[E + e];
    atomicAdd(&agg[(size_t)c * DIM + d], we * g0[(size_t)r * DIM + d]);
  }
}

__global__ __launch_bounds__(256)
void combine_kernel(const float* __restrict__ x2, const float* __restrict__ x1,
                    float* __restrict__ out, size_t total) {
  const size_t idx = (size_t)blockIdx.x * 256 + threadIdx.x;
  if (idx < total) out[idx] = 0.8f * x2[idx] + 0.2f * x1[idx];
}

// ---------------------------------------------------------------------------
extern "C" void kernel_launch(void* const* d_in, const int* in_sizes, int n_in,
                              void* d_out, int out_size, void* d_ws, size_t ws_size,
                              hipStream_t stream) {
  (void)in_sizes; (void)n_in; (void)out_size; (void)ws_size;
  const float* x     = (const float*)d_in[0];
  const int*   ei    = (const int*)d_in[1];
  const float* tW0   = (const float*)d_in[2];
  const float* tb0   = (const float*)d_in[3];
  const float* tln0g = (const float*)d_in[4];
  const float* tln0b = (const float*)d_in[5];
  const float* Wq    = (const float*)d_in[6];
  const float* bq    = (const float*)d_in[7];
  const float* Wk    = (const float*)d_in[8];
  const float* bk    = (const float*)d_in[9];
  const float* Wv    = (const float*)d_in[10];
  const float* bvv   = (const float*)d_in[11];
  const float* tln1g = (const float*)d_in[12];
  const float* tln1b = (const float*)d_in[13];
  const float* gW0   = (const float*)d_in[14];
  const float* gb0   = (const float*)d_in[15];
  const float* gbn0g = (const float*)d_in[16];
  const float* gbn0b = (const float*)d_in[17];
  const float* gWc   = (const float*)d_in[18];
  const float* gbc   = (const float*)d_in[19];
  const float* gbn1g = (const float*)d_in[20];
  const float* gbn1b = (const float*)d_in[21];
  const float* Wo    = (const float*)d_in[22];
  const float* bo    = (const float*)d_in[23];
  float* out = (float*)d_out;

  float* ws = (float*)d_ws;
  const size_t ND = (size_t)N_NODES * DIM;
  float* h    = ws;            // h (later xc)
  float* q    = ws + 1 * ND;
  float* kbuf = ws + 2 * ND;   // k (later x1)
  float* vbuf = ws + 3 * ND;   // v (later x2)
  float* g0   = ws + 4 * ND;
  float* agg  = ws + 5 * ND;
  float* tmp  = ws + 6 * ND;
  float* kvs  = ws + 7 * ND;   // 256x256
  float* ksum = kvs + 65536;
  float* vsum = ksum + 256;
  float* bn0s = vsum + 256;
  float* bn0q = bn0s + 256;
  float* bn1s = bn0q + 256;
  float* bn1q = bn1s + 256;
  float* deg  = bn1q + 256;    // N floats
  float* wbuf = deg + N_NODES; // E floats

  const dim3 blk(256);
  const dim3 g_big(DIM / 64, (N_NODES + 255) / 256);   // 256 rows per block now
  const int CH = 512;
  const unsigned statsBlocks = (N_NODES + CH - 1) / CH;
  const unsigned elemBlocks = (unsigned)((ND + 255) / 256);

  // zero accumulators (agg + [ksum..deg] region) every call (graph-replay safe)
  zero_kernel<<<2048, blk, 0, stream>>>(agg, ND);
  zero_kernel<<<512, blk, 0, stream>>>(ksum, (size_t)(6 * 256 + N_NODES));

  // ---- transformer branch ----
  wmma_gemm_f32<<<g_big, blk, 0, stream>>>(x, DIM, 1, tW0, DIM, tmp, DIM, tb0, N_NODES, DIM);
  ln_relu_kernel<<<N_NODES, blk, 0, stream>>>(tmp, tln0g, tln0b, h);
  wmma_gemm_f32<<<g_big, blk, 0, stream>>>(h, DIM, 1, Wq, DIM, q, DIM, bq, N_NODES, DIM);
  wmma_gemm_f32<<<g_big, blk, 0, stream>>>(h, DIM, 1, Wk, DIM, kbuf, DIM, bk, N_NODES, DIM);
  wmma_gemm_f32<<<g_big, blk, 0, stream>>>(h, DIM, 1, Wv, DIM, vbuf, DIM, bvv, N_NODES, DIM);
  rownorm_kernel<<<N_NODES, blk, 0, stream>>>(q);
  rownorm_kernel<<<N_NODES, blk, 0, stream>>>(kbuf);
  col_stats_kernel<<<statsBlocks, blk, 0, stream>>>(kbuf, ksum, (float*)nullptr, N_NODES, CH);
  col_stats_kernel<<<statsBlocks, blk, 0, stream>>>(vbuf, vsum, (float*)nullptr, N_NODES, CH);
  // kvs = k^T @ v : same GEMM with A strides swapped (rsA=1, csA=DIM), K = N
  wmma_gemm_f32<<<dim3(DIM / 64, 1), blk, 0, stream>>>(kbuf, 1, DIM, vbuf, DIM, kvs, DIM,
                                                       (const float*)nullptr, DIM, N_NODES);
  wmma_gemm_f32<<<g_big, blk, 0, stream>>>(q, DIM, 1, kvs, DIM, tmp, DIM,
                                           (const float*)nullptr, N_NODES, DIM);
  attn_ln_kernel<<<N_NODES, blk, 0, stream>>>(tmp, q, ksum, vsum, h, tln1g, tln1b,
                                              kbuf /*x1*/, (float)N_NODES);

  // ---- gnn branch ----
  wmma_gemm_f32<<<g_big, blk, 0, stream>>>(x, DIM, 1, gW0, DIM, tmp, DIM, gb0, N_NODES, DIM);
  col_stats_kernel<<<statsBlocks, blk, 0, stream>>>(tmp, bn0s, bn0q, N_NODES, CH);
  bn_relu_kernel<<<elemBlocks, blk, 0, stream>>>(tmp, bn0s, bn0q, gbn0g, gbn0b,
                                                 (const float*)nullptr, g0, 1.0f / N_NODES, ND);
  deg_kernel<<<(N_EDGES + 255) / 256, blk, 0, stream>>>(ei, deg, N_EDGES);
  edgew_kernel<<<(N_EDGES + 255) / 256, blk, 0, stream>>>(ei, deg, wbuf, N_EDGES);
  {
    const size_t sc = (size_t)N_EDGES * DIM;
    scatter_kernel<<<(unsigned)((sc + 255) / 256), blk, 0, stream>>>(ei, wbuf, g0, agg, N_EDGES);
  }
  wmma_gemm_f32<<<g_big, blk, 0, stream>>>(agg, DIM, 1, gWc, DIM, tmp, DIM, gbc, N_NODES, DIM);
  col_stats_kernel<<<statsBlocks, blk, 0, stream>>>(tmp, bn1s, bn1q, N_NODES, CH);
  bn_relu_kernel<<<elemBlocks, blk, 0, stream>>>(tmp, bn1s, bn1q, gbn1g, gbn1b,
                                                 g0 /*residual*/, vbuf /*x2*/, 1.0f / N_NODES, ND);

  // ---- combine + head ----
  combine_kernel<<<elemBlocks, blk, 0, stream>>>(vbuf, kbuf, h /*xc*/, ND);
  wmma_gemm_f32<<<dim3(OUTD / 64, (N_NODES + 255) / 256), blk, 0, stream>>>(
      h, DIM, 1, Wo, OUTD, out, OUTD, bo, N_NODES, DIM);
}